// GCN_31190052504412
// MI455X (gfx1250) — compile-verified
//
#include <hip/hip_runtime.h>
#include <hip/hip_bf16.h>

typedef __attribute__((ext_vector_type(2))) float v2f;
typedef __attribute__((ext_vector_type(8))) float v8f;

// ---------------------------------------------------------------------------
// 1) init: deg = 1.0 (self-loop), pooled = 0
// ---------------------------------------------------------------------------
__global__ void k_init(float* __restrict__ deg, float* __restrict__ pooled,
                       int n, int gq) {
    int i = blockIdx.x * blockDim.x + threadIdx.x;
    if (i < n)  deg[i] = 1.0f;
    if (i < gq) pooled[i] = 0.0f;
}

// ---------------------------------------------------------------------------
// 2) degree accumulation over dst
// ---------------------------------------------------------------------------
__global__ void k_degree(const int* __restrict__ dst, float* __restrict__ deg,
                         int e) {
    int i = blockIdx.x * blockDim.x + threadIdx.x;
    if (i < e) atomicAdd(&deg[dst[i]], 1.0f);
}

// ---------------------------------------------------------------------------
// 3) dis = rsqrt(deg), in place
// ---------------------------------------------------------------------------
__global__ void k_rsqrt(float* __restrict__ d, int n) {
    int i = blockIdx.x * blockDim.x + threadIdx.x;
    if (i < n) d[i] = rsqrtf(d[i]);
}

// ---------------------------------------------------------------------------
// 4) layer-1 linear (F_IN=2 -> H=16) + self-loop init of accumulator
//    thread per (node, feature)
// ---------------------------------------------------------------------------
__global__ void k_lin1(const float* __restrict__ x, const float* __restrict__ W1,
                       const float* __restrict__ dis,
                       float* __restrict__ h_lin, float* __restrict__ agg, int n) {
    int i = blockIdx.x * blockDim.x + threadIdx.x;
    if (i >= n * 16) return;
    int node = i >> 4, j = i & 15;
    float h = x[node * 2] * W1[j] + x[node * 2 + 1] * W1[16 + j];
    h_lin[i] = h;
    float dv = dis[node];
    agg[i] = h * dv * dv;
}

// ---------------------------------------------------------------------------
// 5) edge scatter: 16 lanes per edge (lane = feature), coalesced gather+atomic
// ---------------------------------------------------------------------------
__global__ void k_edge(const int* __restrict__ src, const int* __restrict__ dst,
                       const float* __restrict__ dis,
                       const float* __restrict__ h_lin, float* __restrict__ agg,
                       long long total) {
    long long i = (long long)blockIdx.x * blockDim.x + threadIdx.x;
    if (i >= total) return;
    int e = (int)(i >> 4), j = (int)(i & 15);
    int s = src[e], d = dst[e];
    float w = dis[s] * dis[d];
    atomicAdd(&agg[d * 16 + j], h_lin[s * 16 + j] * w);
}

// ---------------------------------------------------------------------------
// 6) bias + relu, in place
// ---------------------------------------------------------------------------
__global__ void k_bias_relu(float* __restrict__ agg, const float* __restrict__ b,
                            int n) {
    int i = blockIdx.x * blockDim.x + threadIdx.x;
    if (i >= n * 16) return;
    float v = agg[i] + b[i & 15];
    agg[i] = v > 0.0f ? v : 0.0f;
}

// ---------------------------------------------------------------------------
// 7) layer-2 GEMM (N x 16) @ (16 x 16) via V_WMMA_F32_16X16X4_F32.
//    One wave per 16-row tile; K=16 chained as four k=4 WMMAs.
//    A layout (16x4 f32): lanes 0-15 hold K={kb,kb+1}, lanes 16-31 K={kb+2,kb+3}.
//    D layout: VGPR v holds (M = v + 8*half, N = lane%16).
//    Uniform tile-level bounds check -> straight-line store path for full tiles.
//    Also writes agg2 init (self-loop term) fused.
// ---------------------------------------------------------------------------
__global__ void k_wmma_l2(const float* __restrict__ h1, const float* __restrict__ W2,
                          const float* __restrict__ dis,
                          float* __restrict__ h2_lin, float* __restrict__ agg2,
                          int n, int ntiles) {
    int wave = (blockIdx.x * blockDim.x + threadIdx.x) >> 5;
    if (wave >= ntiles) return;
    int lane = threadIdx.x & 31;
    int half = lane >> 4, col = lane & 15;
    int row0 = wave * 16;
    int rA = row0 + col;
    if (rA >= n) rA = n - 1;   // only possible in a (nonexistent here) partial tile

    v8f c = {};
#pragma unroll
    for (int kc = 0; kc < 4; ++kc) {
        int kA = kc * 4 + 2 * half;
        v2f a, b;
        a.x = h1[rA * 16 + kA];
        a.y = h1[rA * 16 + kA + 1];
        b.x = W2[kA * 16 + col];
        b.y = W2[(kA + 1) * 16 + col];
        c = __builtin_amdgcn_wmma_f32_16x16x4_f32(false, a, false, b,
                                                  (short)0, c, false, false);
    }

    if (row0 + 16 <= n) {
        // full tile: straight-line stores, no per-row exec-mask branching
#pragma unroll
        for (int v = 0; v < 8; ++v) {
            int rD = row0 + v + 8 * half;
            float val = c[v];
            float dv = dis[rD];
            h2_lin[rD * 16 + col] = val;
            agg2[rD * 16 + col] = val * dv * dv;
        }
    } else {
#pragma unroll
        for (int v = 0; v < 8; ++v) {
            int rD = row0 + v + 8 * half;
            if (rD < n) {
                float val = c[v];
                float dv = dis[rD];
                h2_lin[rD * 16 + col] = val;
                agg2[rD * 16 + col] = val * dv * dv;
            }
        }
    }
}

// ---------------------------------------------------------------------------
// 8) global max pooling via uint atomicMax (valid: inputs >= 0 post-ReLU)
// ---------------------------------------------------------------------------
__global__ void k_pool(const float* __restrict__ h2, const int* __restrict__ batch,
                       unsigned int* __restrict__ pooled_u, int n) {
    int i = blockIdx.x * blockDim.x + threadIdx.x;
    if (i >= n * 16) return;
    int node = i >> 4, j = i & 15;
    atomicMax(&pooled_u[batch[node] * 16 + j], __float_as_uint(h2[i]));
}

// ---------------------------------------------------------------------------
// 9) classifier GEMM (G x 16) @ (16 x 10, zero-padded to 16) via WMMA f32
// ---------------------------------------------------------------------------
__global__ void k_wmma_cls(const float* __restrict__ pooled,
                           const float* __restrict__ Wl, const float* __restrict__ bl,
                           float* __restrict__ logits, int g, int ntiles) {
    int wave = (blockIdx.x * blockDim.x + threadIdx.x) >> 5;
    if (wave >= ntiles) return;
    int lane = threadIdx.x & 31;
    int half = lane >> 4, col = lane & 15;
    int row0 = wave * 16;
    int rA = row0 + col;
    if (rA >= g) rA = g - 1;

    v8f c = {};
#pragma unroll
    for (int kc = 0; kc < 4; ++kc) {
        int kA = kc * 4 + 2 * half;
        v2f a, b;
        a.x = pooled[rA * 16 + kA];
        a.y = pooled[rA * 16 + kA + 1];
        b.x = (col < 10) ? Wl[kA * 10 + col] : 0.0f;
        b.y = (col < 10) ? Wl[(kA + 1) * 10 + col] : 0.0f;
        c = __builtin_amdgcn_wmma_f32_16x16x4_f32(false, a, false, b,
                                                  (short)0, c, false, false);
    }
    float bias = (col < 10) ? bl[col] : 0.0f;

    if (row0 + 16 <= g) {
#pragma unroll
        for (int v = 0; v < 8; ++v) {
            int rD = row0 + v + 8 * half;
            logits[rD * 16 + col] = c[v] + bias;
        }
    } else {
#pragma unroll
        for (int v = 0; v < 8; ++v) {
            int rD = row0 + v + 8 * half;
            if (rD < g) logits[rD * 16 + col] = c[v] + bias;
        }
    }
}

// ---------------------------------------------------------------------------
// 10) per-graph log_softmax over C=10 classes
// ---------------------------------------------------------------------------
__global__ void k_lsm(const float* __restrict__ logits, float* __restrict__ out,
                      int g) {
    int gi = blockIdx.x * blockDim.x + threadIdx.x;
    if (gi >= g) return;
    float m = -3.402823466e+38f;
#pragma unroll
    for (int cidx = 0; cidx < 10; ++cidx) {
        float v = logits[gi * 16 + cidx];
        m = v > m ? v : m;
    }
    float s = 0.0f;
#pragma unroll
    for (int cidx = 0; cidx < 10; ++cidx)
        s += expf(logits[gi * 16 + cidx] - m);
    float ls = logf(s);
#pragma unroll
    for (int cidx = 0; cidx < 10; ++cidx)
        out[gi * 10 + cidx] = logits[gi * 16 + cidx] - m - ls;
}

// ---------------------------------------------------------------------------
extern "C" void kernel_launch(void* const* d_in, const int* in_sizes, int n_in,
                              void* d_out, int out_size, void* d_ws, size_t ws_size,
                              hipStream_t stream) {
    const float* x     = (const float*)d_in[0];
    const int*   eidx  = (const int*)d_in[1];
    const int*   batch = (const int*)d_in[2];
    const float* W1    = (const float*)d_in[3];
    const float* b1    = (const float*)d_in[4];
    const float* W2    = (const float*)d_in[5];
    const float* b2    = (const float*)d_in[6];
    const float* Wl    = (const float*)d_in[7];
    const float* bl    = (const float*)d_in[8];
    float* out = (float*)d_out;

    const int n = in_sizes[0] / 2;        // 500000
    const int e = in_sizes[1] / 2;        // 8000000
    const int g = out_size / 10;          // 1024

    const int* src = eidx;
    const int* dst = eidx + e;

    // workspace layout (256B aligned offsets)
    char* base = (char*)d_ws;
    size_t off = 0;
    float* dis = (float*)(base + off);            off += ((size_t)n * 4 + 255) & ~(size_t)255;
    float* B0  = (float*)(base + off);            off += ((size_t)n * 16 * 4 + 255) & ~(size_t)255; // h1_lin / h2_lin
    float* B1  = (float*)(base + off);            off += ((size_t)n * 16 * 4 + 255) & ~(size_t)255; // agg1/h1 -> agg2/h2
    float* pooled = (float*)(base + off);         off += ((size_t)g * 16 * 4 + 255) & ~(size_t)255;
    float* logits = (float*)(base + off);         off += ((size_t)g * 16 * 4 + 255) & ~(size_t)255;
    (void)ws_size; (void)n_in;

    const int T = 256;
    const long long etot = (long long)e * 16;
    const int nf = n * 16;
    const int ntiles_n = (n + 15) / 16;
    const int ntiles_g = (g + 15) / 16;

    // normalization
    k_init  <<<(n + T - 1) / T, T, 0, stream>>>(dis, pooled, n, g * 16);
    k_degree<<<(e + T - 1) / T, T, 0, stream>>>(dst, dis, e);
    k_rsqrt <<<(n + T - 1) / T, T, 0, stream>>>(dis, n);

    // layer 1
    k_lin1     <<<(nf + T - 1) / T, T, 0, stream>>>(x, W1, dis, B0, B1, n);
    k_edge     <<<(unsigned)((etot + T - 1) / T), T, 0, stream>>>(src, dst, dis, B0, B1, etot);
    k_bias_relu<<<(nf + T - 1) / T, T, 0, stream>>>(B1, b1, n);

    // layer 2 (WMMA GEMM; writes h2_lin into B0, agg2 self-loop init into B1)
    k_wmma_l2  <<<(ntiles_n * 32 + T - 1) / T, T, 0, stream>>>(B1, W2, dis, B0, B1, n, ntiles_n);
    k_edge     <<<(unsigned)((etot + T - 1) / T), T, 0, stream>>>(src, dst, dis, B0, B1, etot);
    k_bias_relu<<<(nf + T - 1) / T, T, 0, stream>>>(B1, b2, n);

    // pooling + classifier + log_softmax
    k_pool    <<<(nf + T - 1) / T, T, 0, stream>>>(B1, batch, (unsigned int*)pooled, n);
    k_wmma_cls<<<(ntiles_g * 32 + T - 1) / T, T, 0, stream>>>(pooled, Wl, bl, logits, g, ntiles_g);
    k_lsm     <<<(g + T - 1) / T, T, 0, stream>>>(logits, out, g);
}